// MolecularGCNModel_11897059410632
// MI455X (gfx1250) — compile-verified
//
#include <hip/hip_runtime.h>
#include <hip/hip_bf16.h>
#include <cstddef>

// ---------------------------------------------------------------------------
// MolecularGCNModel for MI455X (gfx1250, wave32).
// GEMMs use V_WMMA_F32_16X16X4_F32 (full-precision f32 matrix op) so the
// matrix pipe is used without precision loss vs the f32 reference.
// W is staged in LDS *k-pair packed* so each B fragment is a single aligned
// ds_load_b64 straight into an even VGPR pair (no repack movs before WMMA).
// Edge aggregation (the HBM-roofline-dominant part, ~5-6 GB irregular traffic)
// uses wave-per-edge float2 gathers + global f32 atomics; feature matrices
// (25.6MB) fit in the 192MB L2 so the scatter mostly hits L2.
// ---------------------------------------------------------------------------

#define N_NODES 100000
#define N_EDGES 3200000
#define N_GRAPHS 256
#define IN_DIM 128
#define HID 64
#define BN_EPS 1e-5f

typedef __attribute__((ext_vector_type(2))) float v2f;
typedef __attribute__((ext_vector_type(8))) float v8f;

// ---------------------------------------------------------------------------
// Utility kernels
// ---------------------------------------------------------------------------
__global__ void zero_kernel(float* __restrict__ p, int n) {
    int i = blockIdx.x * blockDim.x + threadIdx.x;
    if (i < n) p[i] = 0.0f;
}

__global__ void deg_init_kernel(float* __restrict__ deg, int n) {
    int i = blockIdx.x * blockDim.x + threadIdx.x;
    if (i < n) deg[i] = 1.0f;  // self loop
}

__global__ void deg_count_kernel(const int* __restrict__ dst, float* __restrict__ deg, int nE) {
    int i = blockIdx.x * blockDim.x + threadIdx.x;
    if (i < nE) atomicAdd(&deg[dst[i]], 1.0f);
}

__global__ void deg_inv_kernel(float* __restrict__ deg, int n) {
    int i = blockIdx.x * blockDim.x + threadIdx.x;
    if (i < n) deg[i] = rsqrtf(deg[i]);  // in place -> dinv
}

// ---------------------------------------------------------------------------
// WMMA GEMM: Out[M x 64] = X[M x K] @ W[K x 64]
// One wave computes a 16x64 tile (4 accumulators of v8f).
//
// Fragment layouts per CDNA5 ISA (wave32):
//   A (16x4 f32):  lane L<16 -> row M=L, holds K=k+0 (v0), K=k+1 (v1);
//                  lane L>=16 -> row M=L-16, holds K=k+2 (v0), K=k+3 (v1).
//   B (4x16 f32):  lane L<16 -> col N=L,  K=k+0/k+1; lane>=16 -> col L-16, K=k+2/k+3.
//   C/D (16x16):   VGPR r, lane<16 -> (M=r, N=lane); lane>=16 -> (M=r+8, N=lane-16).
//
// LDS holds W packed by k-pairs: sWp[p*64 + col] = (W[2p][col], W[2p+1][col]);
// a lane's B fragment for k-step k is sWp[(k/2 + hiHalf)*64 + col]: one b64 load.
// ---------------------------------------------------------------------------
template <int K>
__global__ void gemm_wmma_kernel(const float* __restrict__ X,
                                 const float* __restrict__ W,
                                 float* __restrict__ Out, int M) {
    __shared__ __align__(16) float sW[K * 64];
    v2f* sWp = (v2f*)sW;
    for (int i = threadIdx.x; i < (K / 2) * 64; i += blockDim.x) {
        int p = i >> 6;
        int col = i & 63;
        v2f pr;
        pr.x = W[(size_t)(2 * p) * 64 + col];
        pr.y = W[(size_t)(2 * p + 1) * 64 + col];
        sWp[i] = pr;
    }
    __syncthreads();

    const int wave   = threadIdx.x >> 5;
    const int lane   = threadIdx.x & 31;
    const int nWaves = blockDim.x >> 5;
    const int rowTile = blockIdx.x * nWaves + wave;
    if (rowTile * 16 >= M) return;  // wave-uniform: EXEC stays all-ones for WMMA

    const int row0   = rowTile * 16;
    const int mLocal = lane & 15;        // A row / B,C column within tile
    const int hiHalf = (lane >> 4);      // 0: k+0/k+1 & C rows 0..7; 1: k+2/k+3 & C rows 8..15
    const int kb     = hiHalf * 2;       // k sub-offset for the A fragment

    v8f c0 = {}, c1 = {}, c2 = {}, c3 = {};
    const float* xrow = X + (size_t)(row0 + mLocal) * K;

#pragma unroll 4
    for (int k = 0; k < K; k += 4) {
        v2f a = *(const v2f*)(xrow + k + kb);                       // b64 global load
        const v2f* brow = sWp + (size_t)(k / 2 + hiHalf) * 64 + mLocal;
        v2f b0 = brow[0];                                           // b64 LDS loads
        v2f b1 = brow[16];
        v2f b2 = brow[32];
        v2f b3 = brow[48];
        c0 = __builtin_amdgcn_wmma_f32_16x16x4_f32(false, a, false, b0, (short)0, c0, false, false);
        c1 = __builtin_amdgcn_wmma_f32_16x16x4_f32(false, a, false, b1, (short)0, c1, false, false);
        c2 = __builtin_amdgcn_wmma_f32_16x16x4_f32(false, a, false, b2, (short)0, c2, false, false);
        c3 = __builtin_amdgcn_wmma_f32_16x16x4_f32(false, a, false, b3, (short)0, c3, false, false);
    }

#pragma unroll
    for (int r = 0; r < 8; ++r) {
        float* orow = Out + (size_t)(row0 + r + hiHalf * 8) * 64 + mLocal;
        orow[0]  = c0[r];
        orow[16] = c1[r];
        orow[32] = c2[r];
        orow[48] = c3[r];
    }
}

// ---------------------------------------------------------------------------
// agg = dinv^2 * h + bias   (self-loop term; also initializes the scatter dest)
// ---------------------------------------------------------------------------
__global__ void agg_init_kernel(const float* __restrict__ h,
                                const float* __restrict__ dinv,
                                const float* __restrict__ bias,
                                float* __restrict__ agg, int n) {
    int i = blockIdx.x * blockDim.x + threadIdx.x;
    if (i >= n) return;
    int node = i >> 6;
    int f = i & 63;
    float di = dinv[node];
    agg[i] = di * di * h[i] + bias[f];
}

// ---------------------------------------------------------------------------
// Wave-per-edge scatter: agg[dst] += dinv[src]*dinv[dst] * h[src]
// Each lane handles 2 of the 64 features (float2 gather, 2 f32 atomics).
// ---------------------------------------------------------------------------
__global__ void edge_scatter_kernel(const int* __restrict__ src,
                                    const int* __restrict__ dst,
                                    const float* __restrict__ dinv,
                                    const float* __restrict__ h,
                                    float* __restrict__ agg, int nE) {
    int wave = (int)((blockIdx.x * blockDim.x + threadIdx.x) >> 5);
    int lane = threadIdx.x & 31;
    if (wave >= nE) return;
    int s = src[wave];
    int d = dst[wave];
    float coef = dinv[s] * dinv[d];
    float2 v = ((const float2*)(h + (size_t)s * 64))[lane];
    float* ad = agg + (size_t)d * 64 + lane * 2;
    atomicAdd(ad + 0, v.x * coef);
    atomicAdd(ad + 1, v.y * coef);
}

// ---------------------------------------------------------------------------
// BatchNorm statistics: per-column sum and sum-of-squares (64 columns).
// Block-local LDS reduction, then 128 global atomics per block.
// ---------------------------------------------------------------------------
__global__ void bn_stats_kernel(const float* __restrict__ x,
                                float* __restrict__ sums,
                                float* __restrict__ sumsq, int nRows) {
    const int c = threadIdx.x & 63;
    const int g = threadIdx.x >> 6;                 // 0..3 row groups per block
    const int rowStart = blockIdx.x * (blockDim.x >> 6) + g;
    const int stride   = gridDim.x * (blockDim.x >> 6);
    float s = 0.0f, q = 0.0f;
    for (int r = rowStart; r < nRows; r += stride) {
        float v = x[(size_t)r * 64 + c];
        s += v;
        q += v * v;
    }
    __shared__ float ls[256], lq[256];
    ls[threadIdx.x] = s;
    lq[threadIdx.x] = q;
    __syncthreads();
    if (threadIdx.x < 64) {
        s = ls[threadIdx.x] + ls[threadIdx.x + 64] + ls[threadIdx.x + 128] + ls[threadIdx.x + 192];
        q = lq[threadIdx.x] + lq[threadIdx.x + 64] + lq[threadIdx.x + 128] + lq[threadIdx.x + 192];
        atomicAdd(&sums[threadIdx.x], s);
        atomicAdd(&sumsq[threadIdx.x], q);
    }
}

// ---------------------------------------------------------------------------
// Fused normalize + gamma/beta + optional residual + ReLU
// ---------------------------------------------------------------------------
__global__ void bn_apply_kernel(const float* __restrict__ x,
                                const float* __restrict__ sums,
                                const float* __restrict__ sumsq,
                                const float* __restrict__ gamma,
                                const float* __restrict__ beta,
                                const float* __restrict__ res,
                                float* __restrict__ out, int n, float invN) {
    int i = blockIdx.x * blockDim.x + threadIdx.x;
    if (i >= n) return;
    int f = i & 63;
    float mu  = sums[f] * invN;
    float var = sumsq[f] * invN - mu * mu;
    float v = (x[i] - mu) * rsqrtf(var + BN_EPS) * gamma[f] + beta[f];
    if (res) v += res[i];
    out[i] = fmaxf(v, 0.0f);
}

// ---------------------------------------------------------------------------
// Global mean pool: wave-per-node scatter of features + count.
// ---------------------------------------------------------------------------
__global__ void pool_scatter_kernel(const float* __restrict__ h,
                                    const int* __restrict__ batch,
                                    float* __restrict__ psum,
                                    float* __restrict__ pcnt, int nNodes) {
    int wave = (int)((blockIdx.x * blockDim.x + threadIdx.x) >> 5);
    int lane = threadIdx.x & 31;
    if (wave >= nNodes) return;
    int g = batch[wave];
    float2 v = ((const float2*)(h + (size_t)wave * 64))[lane];
    float* pd = psum + (size_t)g * 64 + lane * 2;
    atomicAdd(pd + 0, v.x);
    atomicAdd(pd + 1, v.y);
    if (lane == 0) atomicAdd(&pcnt[g], 1.0f);
}

// ---------------------------------------------------------------------------
// Final FC: one wave per graph; dot(pooled_mean, fcW) + fcb via wave32 shuffle.
// ---------------------------------------------------------------------------
__global__ void final_fc_kernel(const float* __restrict__ psum,
                                const float* __restrict__ pcnt,
                                const float* __restrict__ fcW,
                                const float* __restrict__ fcb,
                                float* __restrict__ out, int nGraphs) {
    int wave = (int)((blockIdx.x * blockDim.x + threadIdx.x) >> 5);
    int lane = threadIdx.x & 31;
    if (wave >= nGraphs) return;
    float invc = 1.0f / fmaxf(pcnt[wave], 1.0f);
    float2 p = ((const float2*)(psum + (size_t)wave * 64))[lane];
    float2 w = ((const float2*)fcW)[lane];
    float partial = (p.x * w.x + p.y * w.y) * invc;
#pragma unroll
    for (int off = 16; off > 0; off >>= 1) partial += __shfl_down(partial, off, 32);
    if (lane == 0) out[wave] = partial + fcb[0];
}

// ---------------------------------------------------------------------------
// Host launcher
// ---------------------------------------------------------------------------
extern "C" void kernel_launch(void* const* d_in, const int* in_sizes, int n_in,
                              void* d_out, int out_size, void* d_ws, size_t ws_size,
                              hipStream_t stream) {
    (void)in_sizes; (void)n_in; (void)out_size; (void)ws_size;

    const float* x    = (const float*)d_in[0];
    const int*   ei   = (const int*)d_in[1];       // [2, E] row-major
    const int*   batch= (const int*)d_in[2];
    const float* W1   = (const float*)d_in[3];
    const float* b1   = (const float*)d_in[4];
    const float* g1   = (const float*)d_in[5];
    const float* be1  = (const float*)d_in[6];
    const float* W2   = (const float*)d_in[7];
    const float* b2   = (const float*)d_in[8];
    const float* g2   = (const float*)d_in[9];
    const float* be2  = (const float*)d_in[10];
    const float* W3   = (const float*)d_in[11];
    const float* b3   = (const float*)d_in[12];
    const float* g3   = (const float*)d_in[13];
    const float* be3  = (const float*)d_in[14];
    const float* fcW  = (const float*)d_in[15];
    const float* fcb  = (const float*)d_in[16];
    float* out = (float*)d_out;

    const int* src = ei;
    const int* dst = ei + N_EDGES;

    // workspace layout (floats); all offsets 16B-aligned
    float* ws    = (float*)d_ws;
    float* dinv  = ws;                       // N_NODES
    float* bufA  = dinv + N_NODES;           // N_NODES*64
    float* bufB  = bufA + (size_t)N_NODES * 64;
    float* bufC  = bufB + (size_t)N_NODES * 64;
    float* sums  = bufC + (size_t)N_NODES * 64;   // 64
    float* sumsq = sums + 64;                     // 64
    float* psum  = sumsq + 64;                    // N_GRAPHS*64
    float* pcnt  = psum + (size_t)N_GRAPHS * 64;  // N_GRAPHS

    const int T = 256;
    const int nFeat = N_NODES * HID;
    const float invN = 1.0f / (float)N_NODES;

    // degrees -> dinv (in place)
    deg_init_kernel<<<(N_NODES + T - 1) / T, T, 0, stream>>>(dinv, N_NODES);
    deg_count_kernel<<<(N_EDGES + T - 1) / T, T, 0, stream>>>(dst, dinv, N_EDGES);
    deg_inv_kernel<<<(N_NODES + T - 1) / T, T, 0, stream>>>(dinv, N_NODES);

    const int rowTiles = N_NODES / 16;                 // 6250 (exact)
    const int gemmBlocks = (rowTiles + (T / 32) - 1) / (T / 32);
    const int edgeBlocks = (int)(((size_t)N_EDGES * 32 + T - 1) / T);
    const int nodeWaveBlocks = (int)(((size_t)N_NODES * 32 + T - 1) / T);

    // ---------------- Layer 1 ----------------
    gemm_wmma_kernel<IN_DIM><<<gemmBlocks, T, 0, stream>>>(x, W1, bufA, N_NODES);
    agg_init_kernel<<<(nFeat + T - 1) / T, T, 0, stream>>>(bufA, dinv, b1, bufB, nFeat);
    edge_scatter_kernel<<<edgeBlocks, T, 0, stream>>>(src, dst, dinv, bufA, bufB, N_EDGES);
    zero_kernel<<<1, 128, 0, stream>>>(sums, 128);
    bn_stats_kernel<<<400, T, 0, stream>>>(bufB, sums, sumsq, N_NODES);
    bn_apply_kernel<<<(nFeat + T - 1) / T, T, 0, stream>>>(bufB, sums, sumsq, g1, be1,
                                                           nullptr, bufA, nFeat, invN);
    // bufA = h1 (also the residual)

    // ---------------- Layer 2 ----------------
    gemm_wmma_kernel<HID><<<gemmBlocks, T, 0, stream>>>(bufA, W2, bufC, N_NODES);
    agg_init_kernel<<<(nFeat + T - 1) / T, T, 0, stream>>>(bufC, dinv, b2, bufB, nFeat);
    edge_scatter_kernel<<<edgeBlocks, T, 0, stream>>>(src, dst, dinv, bufC, bufB, N_EDGES);
    zero_kernel<<<1, 128, 0, stream>>>(sums, 128);
    bn_stats_kernel<<<400, T, 0, stream>>>(bufB, sums, sumsq, N_NODES);
    bn_apply_kernel<<<(nFeat + T - 1) / T, T, 0, stream>>>(bufB, sums, sumsq, g2, be2,
                                                           bufA, bufC, nFeat, invN);
    // bufC = h2

    // ---------------- Layer 3 ----------------
    gemm_wmma_kernel<HID><<<gemmBlocks, T, 0, stream>>>(bufC, W3, bufA, N_NODES);
    agg_init_kernel<<<(nFeat + T - 1) / T, T, 0, stream>>>(bufA, dinv, b3, bufB, nFeat);
    edge_scatter_kernel<<<edgeBlocks, T, 0, stream>>>(src, dst, dinv, bufA, bufB, N_EDGES);
    zero_kernel<<<1, 128, 0, stream>>>(sums, 128);
    bn_stats_kernel<<<400, T, 0, stream>>>(bufB, sums, sumsq, N_NODES);
    bn_apply_kernel<<<(nFeat + T - 1) / T, T, 0, stream>>>(bufB, sums, sumsq, g3, be3,
                                                           nullptr, bufA, nFeat, invN);
    // bufA = h3

    // ---------------- Pool + FC ----------------
    zero_kernel<<<(N_GRAPHS * 64 + N_GRAPHS + T - 1) / T, T, 0, stream>>>(psum, N_GRAPHS * 64 + N_GRAPHS);
    pool_scatter_kernel<<<nodeWaveBlocks, T, 0, stream>>>(bufA, batch, psum, pcnt, N_NODES);
    final_fc_kernel<<<(N_GRAPHS * 32 + T - 1) / T, T, 0, stream>>>(psum, pcnt, fcW, fcb, out, N_GRAPHS);
}